// SpconvNet_59390807769140
// MI455X (gfx1250) — compile-verified
//
#include <hip/hip_runtime.h>

// ---------------- problem constants (from reference) ----------------
#define N_VOX   120000
#define IN_CH   32
#define OUT_CH  64
#define NKV     27            // 3^3 kernel offsets
#define DZ      21            // out depth  = (41-1)/2+1
#define HY      200           // out height = (400-1)/2+1
#define WX      176           // out width  = (352-1)/2+1
#define GDUMP   (2*DZ*HY*WX)  // 1,478,400 : dump slot index ("invalid")
#define NTILES  (N_VOX/16)    // 7500 tiles of 16 voxels

// LDS row strides chosen so a wave's 32 x ds_read_b64 hits all 64 banks once:
// (36*n mod 64) and (68*n mod 64) for n=0..15 give disjoint 4-word groups.
#define STRIDE1 36            // padded row stride for 32-wide K rows (phase 1)
#define STRIDE2 68            // padded row stride for 64-wide K rows (phase 2)

typedef __attribute__((ext_vector_type(2))) float v2f;
typedef __attribute__((ext_vector_type(8))) float v8f;

// D(16x16,f32) = A(16x4,f32) x B(4x16,f32) + C  -- native f32 WMMA (CDNA5)
__device__ __forceinline__ v8f wmma4(v2f a, v2f b, v8f c) {
    return __builtin_amdgcn_wmma_f32_16x16x4_f32(
        false, a, false, b, (short)0, c, false, false);
}

// Async-stage one [rows x rowlen] f32 weight slice global->LDS with row padding.
// 512 x 16B chunks, 2 per thread (256-thread block). Tracked by ASYNCcnt.
template <int ROWLEN, int STRIDE>
__device__ __forceinline__ void stage_w(float* dst_lds, const float* src, int tid) {
#pragma unroll
    for (int q = 0; q < 2; ++q) {
        int idx = tid + q * 256;
        int row = idx / (ROWLEN / 4);
        int c   = idx % (ROWLEN / 4);
        unsigned lds = (unsigned)(uintptr_t)(dst_lds + row * STRIDE + c * 4);
        unsigned long long g =
            (unsigned long long)(uintptr_t)(src + row * ROWLEN + c * 4);
        asm volatile("global_load_async_to_lds_b128 %0, %1, off"
                     :: "v"(lds), "v"(g) : "memory");
    }
}
__device__ __forceinline__ void wait_async_le2() {
    asm volatile("s_wait_asynccnt 0x2" ::: "memory");
}
__device__ __forceinline__ void wait_async_0() {
    asm volatile("s_wait_asynccnt 0x0" ::: "memory");
}

// ---------------- kernel 0: transpose weights to [kv][n][k] ----------------
__global__ void k_transpose_w(const float* __restrict__ w1,
                              const float* __restrict__ w2,
                              float* __restrict__ w1t,
                              float* __restrict__ w2t) {
    int tid = blockIdx.x * blockDim.x + threadIdx.x;
    const int total = NKV * IN_CH * OUT_CH;          // 55296 per tensor
    if (tid < total) {
        int kv = tid / 2048, r = tid % 2048;
        int k = r / 64, n = r % 64;                  // W1[kv][k][n]
        w1t[kv * 2048 + n * 32 + k] = w1[tid];
    } else if (tid < 2 * total) {
        int t = tid - total;
        int kv = t / 2048, r = t % 2048;
        int k = r / 32, n = r % 32;                  // W2[kv][k][n]
        w2t[kv * 2048 + n * 64 + k] = w2[t];
    }
}

// ---------------- kernel 1: rulebook + zero touched grid rows ----------------
__global__ void k_rulebook(const int4* __restrict__ coors,
                           int* __restrict__ lins,
                           float* __restrict__ grid) {
    int i = blockIdx.x * blockDim.x + threadIdx.x;
    if (i < 64) grid[(size_t)GDUMP * 64 + i] = 0.0f;   // zero dump row
    if (i >= N_VOX) return;
    int4 c = coors[i];                                  // (b, z, y, x)
#pragma unroll
    for (int kv = 0; kv < NKV; ++kv) {
        int oz = kv / 9, oy = (kv / 3) % 3, ox = kv % 3;
        int cz = c.y + 1 - oz, cy = c.z + 1 - oy, cx = c.w + 1 - ox;
        int lin = GDUMP;
        int all = cz | cy | cx;                         // sign bit: any < 0
        if (all >= 0 && !(all & 1)) {                   // all >=0 and all even
            int qz = cz >> 1, qy = cy >> 1, qx = cx >> 1;
            if (qz < DZ && qy < HY && qx < WX)
                lin = ((c.x * DZ + qz) * HY + qy) * WX + qx;
        }
        lins[kv * N_VOX + i] = lin;
        if (lin != GDUMP) {                             // zero only touched rows
            float4 zv = make_float4(0.f, 0.f, 0.f, 0.f);
            float4* row = (float4*)(grid + (size_t)lin * 64);
#pragma unroll
            for (int j = 0; j < 16; ++j) row[j] = zv;
        }
    }
}

// ---------------- kernel 2: phase 1  grid[lin] += feat @ W1[kv] ----------------
// 8 waves/block, one 16-voxel tile per wave. Weights double-buffered in LDS via
// async global->LDS DMA; 27 x (8 K-steps x 4 N-tiles) f32 WMMA; atomic scatter.
// lins lookups are software-pipelined one kv ahead to hide load latency.
__global__ void __launch_bounds__(256)
k_phase1(const float* __restrict__ feat,
         const float* __restrict__ w1t,
         const int* __restrict__ lins,
         float* __restrict__ grid) {
    __shared__ float sb[2][64 * STRIDE1];               // 2 x 9216B
    int tid  = threadIdx.x;
    int wave = (blockIdx.x * blockDim.x + tid) >> 5;
    bool active = wave < NTILES;                        // wave-uniform
    int lane = tid & 31;
    int m16 = lane & 15, hi = lane >> 4;
    int base = wave * 16;

    // A fragments: 16x32 f32 features tile. 16x4 f32 A layout: lanes 0-15 hold
    // K=(4s,4s+1) of row m16, lanes 16-31 hold K=(4s+2,4s+3).
    v2f a[8];
    int lin = GDUMP;
    if (active) {
        const float* arow = feat + (size_t)(base + m16) * IN_CH + 2 * hi;
#pragma unroll
        for (int s = 0; s < 8; ++s) a[s] = *(const v2f*)(arow + 4 * s);
        lin = lins[base + m16];                         // kv = 0 index
    }

    stage_w<32, STRIDE1>(sb[0], w1t, tid);              // prologue: kv = 0

    for (int kv = 0; kv < NKV; ++kv) {
        if (kv + 1 < NKV) {
            stage_w<32, STRIDE1>(sb[(kv + 1) & 1], w1t + (kv + 1) * 2048, tid);
            wait_async_le2();                           // kv's 2 chunks landed
        } else {
            wait_async_0();
        }
        __syncthreads();                                // slice kv visible to all

        int lin_next = GDUMP;                           // prefetch next index
        if (active && kv + 1 < NKV)
            lin_next = lins[(kv + 1) * N_VOX + base + m16];

        if (active && __any(lin != GDUMP)) {            // wave-uniform skip
            // Batch all 32 B fragments first so the scheduler can pipeline
            // ds loads against the WMMA chain (waits with decreasing DScnt).
            const float* wb = sb[kv & 1] + m16 * STRIDE1 + 2 * hi;
            v2f bf[32];
#pragma unroll
            for (int s = 0; s < 8; ++s) {
#pragma unroll
                for (int nt = 0; nt < 4; ++nt)
                    bf[nt * 8 + s] =
                        *(const v2f*)(wb + nt * 16 * STRIDE1 + 4 * s);
            }
            v8f c0 = {}, c1 = {}, c2 = {}, c3 = {};
#pragma unroll
            for (int s = 0; s < 8; ++s) {
                c0 = wmma4(a[s], bf[0 * 8 + s], c0);
                c1 = wmma4(a[s], bf[1 * 8 + s], c1);
                c2 = wmma4(a[s], bf[2 * 8 + s], c2);
                c3 = wmma4(a[s], bf[3 * 8 + s], c3);
            }
            // Scatter C: element (row m = v+8*hi, col n = nt*16+m16).
#pragma unroll
            for (int v = 0; v < 8; ++v) {
                int m = v + 8 * hi;
                int linm = __shfl(lin, m, 32);
                if (linm != GDUMP) {
                    float* gr = grid + (size_t)linm * 64 + m16;
                    unsafeAtomicAdd(gr + 0,  c0[v]);    // global_atomic_add_f32
                    unsafeAtomicAdd(gr + 16, c1[v]);
                    unsafeAtomicAdd(gr + 32, c2[v]);
                    unsafeAtomicAdd(gr + 48, c3[v]);
                }
            }
        }
        __syncthreads();                                // protect buffer reuse
        lin = lin_next;
    }
}

// ---------------- kernel 3: phase 2  out = sum_kv grid[lin] @ W2[kv] ----------------
__global__ void __launch_bounds__(256)
k_phase2(const float* __restrict__ grid,
         const float* __restrict__ w2t,
         const int* __restrict__ lins,
         float* __restrict__ out) {
    __shared__ float sb[2][32 * STRIDE2];               // 2 x 8704B
    int tid  = threadIdx.x;
    int wave = (blockIdx.x * blockDim.x + tid) >> 5;
    bool active = wave < NTILES;
    int lane = tid & 31;
    int m16 = lane & 15, hi = lane >> 4;
    int base = wave * 16;

    v8f acc0 = {}, acc1 = {};
    int lin = GDUMP;
    if (active) lin = lins[base + m16];                 // kv = 0 index

    stage_w<64, STRIDE2>(sb[0], w2t, tid);              // prologue: kv = 0

    for (int kv = 0; kv < NKV; ++kv) {
        if (kv + 1 < NKV) {
            stage_w<64, STRIDE2>(sb[(kv + 1) & 1], w2t + (kv + 1) * 2048, tid);
            wait_async_le2();
        } else {
            wait_async_0();
        }
        __syncthreads();

        int lin_next = GDUMP;
        if (active && kv + 1 < NKV)
            lin_next = lins[(kv + 1) * N_VOX + base + m16];

        if (active && __any(lin != GDUMP)) {            // invalid rows read zero row
            // Issue all global gathers first, then all LDS loads, then WMMA.
            const float* ar = grid + (size_t)lin * 64 + 2 * hi;
            v2f av[16], bf[32];
#pragma unroll
            for (int s = 0; s < 16; ++s) av[s] = *(const v2f*)(ar + 4 * s);
            const float* wb = sb[kv & 1] + m16 * STRIDE2 + 2 * hi;
#pragma unroll
            for (int s = 0; s < 16; ++s) {
                bf[s]      = *(const v2f*)(wb + 4 * s);
                bf[16 + s] = *(const v2f*)(wb + 16 * STRIDE2 + 4 * s);
            }
#pragma unroll
            for (int s = 0; s < 16; ++s) {              // K = 64
                acc0 = wmma4(av[s], bf[s], acc0);
                acc1 = wmma4(av[s], bf[16 + s], acc1);
            }
        }
        __syncthreads();
        lin = lin_next;
    }

    if (active) {                                        // C: out[(base+m)*32+n]
#pragma unroll
        for (int v = 0; v < 8; ++v) {
            int m = base + v + 8 * hi;
            out[(size_t)m * IN_CH + m16]      = acc0[v];
            out[(size_t)m * IN_CH + 16 + m16] = acc1[v];
        }
    }
}

// ---------------- launcher ----------------
extern "C" void kernel_launch(void* const* d_in, const int* in_sizes, int n_in,
                              void* d_out, int out_size, void* d_ws, size_t ws_size,
                              hipStream_t stream) {
    const float* feat  = (const float*)d_in[0];   // (120000,32) f32
    const int4*  coors = (const int4*) d_in[1];   // (120000,4)  i32
    const float* w1    = (const float*)d_in[2];   // (27,32,64)  f32
    const float* w2    = (const float*)d_in[3];   // (27,64,32)  f32
    float* out = (float*)d_out;                   // (120000,32) f32

    // workspace layout (16B aligned blocks)
    char* ws = (char*)d_ws;
    const size_t lins_bytes = (size_t)NKV * N_VOX * sizeof(int);   // 12,960,000
    const size_t wt_bytes   = (size_t)NKV * IN_CH * OUT_CH * 4;    // 221,184
    int*   lins = (int*)ws;
    float* w1t  = (float*)(ws + lins_bytes);
    float* w2t  = (float*)(ws + lins_bytes + wt_bytes);
    float* grid = (float*)(ws + lins_bytes + 2 * wt_bytes);        // (G+1)*64 f32

    {   // weight transpose
        int threads = 2 * NKV * IN_CH * OUT_CH;
        k_transpose_w<<<(threads + 255) / 256, 256, 0, stream>>>(w1, w2, w1t, w2t);
    }
    {   // rulebook + sparse zeroing of touched grid rows
        k_rulebook<<<(N_VOX + 255) / 256, 256, 0, stream>>>(coors, lins, grid);
    }
    {   // phase 1: async-LDS weight staging + WMMA + atomic scatter
        int blocks = (NTILES + 7) / 8;
        k_phase1<<<blocks, 256, 0, stream>>>(feat, w1t, lins, grid);
    }
    {   // phase 2: gather + async-LDS weight staging + WMMA accumulate
        int blocks = (NTILES + 7) / 8;
        k_phase2<<<blocks, 256, 0, stream>>>(grid, w2t, lins, out);
    }
}